// CBAM_35828617183344
// MI455X (gfx1250) — compile-verified
//
#include <hip/hip_runtime.h>
#include <hip/hip_bf16.h>

typedef __attribute__((ext_vector_type(2))) float v2f;
typedef __attribute__((ext_vector_type(8))) float v8f;

#define NEG_INF (-3.402823466e+38f)

// ---------------------------------------------------------------------------
// Kernel 1: per-(b,c) avg & max over the 3136 contiguous pixels.
// Writes stacked pooled matrix V[64][512]: rows 0..31 = avg, rows 32..63 = max.
// ---------------------------------------------------------------------------
__global__ void __launch_bounds__(256) k_channel_pool(const float* __restrict__ x,
                                                      float* __restrict__ V) {
    const int row = blockIdx.x;                     // b*512 + c
    const float4* p = reinterpret_cast<const float4*>(x + (size_t)row * 3136);
    float s = 0.0f, m = NEG_INF;
    for (int i = threadIdx.x; i < 784; i += 256) {  // 784 float4 = 3136 floats
        float4 v = p[i];
        s += v.x + v.y + v.z + v.w;
        m = fmaxf(m, fmaxf(fmaxf(v.x, v.y), fmaxf(v.z, v.w)));
    }
    // wave32 reduction
    for (int off = 16; off > 0; off >>= 1) {
        s += __shfl_down(s, off);
        m = fmaxf(m, __shfl_down(m, off));
    }
    __shared__ float ss[8], sm[8];
    const int wave = threadIdx.x >> 5, lane = threadIdx.x & 31;
    if (lane == 0) { ss[wave] = s; sm[wave] = m; }
    __syncthreads();
    if (threadIdx.x == 0) {
        float ts = 0.0f, tm = NEG_INF;
        for (int w = 0; w < 8; ++w) { ts += ss[w]; tm = fmaxf(tm, sm[w]); }
        const int b = row >> 9, c = row & 511;
        V[b * 512 + c]        = ts * (1.0f / 3136.0f);   // avg rows
        V[(b + 32) * 512 + c] = tm;                      // max rows
    }
}

// ---------------------------------------------------------------------------
// Kernel 2: channel-attention MLP with fp32 WMMA (exact vs reference).
//   H = relu(V[64x512] @ w1^T)   (w1: [32,512])
//   Y = H[64x32]   @ w2^T        (w2: [512,32])
//   att[b][c] = sigmoid(Y[b][c] + Y[b+32][c])
// One workgroup, 8 waves. A frag: lane->M, half/VGPR->K. B frag: lane->N.
// ---------------------------------------------------------------------------
__global__ void __launch_bounds__(256) k_mlp_att(const float* __restrict__ V,
                                                 const float* __restrict__ w1,
                                                 const float* __restrict__ w2,
                                                 float* __restrict__ att) {
    __shared__ float Hs[64 * 32];
    const int wave = threadIdx.x >> 5;
    const int lane = threadIdx.x & 31;
    const int l16  = lane & 15;
    const int half = lane >> 4;                     // 0: lanes 0-15, 1: lanes 16-31

    // ---- GEMM1: 4 M-tiles x 2 N-tiles = 8 tiles, one per wave; K = 512 ----
    {
        const int mt = wave >> 1;                   // 0..3
        const int nt = wave & 1;                    // 0..1
        v8f acc = {};
        const float* Arow = V  + (size_t)(mt * 16 + l16) * 512;
        const float* Brow = w1 + (size_t)(nt * 16 + l16) * 512;
        for (int k0 = 0; k0 < 512; k0 += 4) {
            v2f a, b;
            a.x = Arow[k0 + 2 * half + 0];
            a.y = Arow[k0 + 2 * half + 1];
            b.x = Brow[k0 + 2 * half + 0];
            b.y = Brow[k0 + 2 * half + 1];
            acc = __builtin_amdgcn_wmma_f32_16x16x4_f32(false, a, false, b,
                                                        (short)0, acc, false, false);
        }
        for (int j = 0; j < 8; ++j) {
            const int m = mt * 16 + j + 8 * half;
            const int n = nt * 16 + l16;
            Hs[m * 32 + n] = fmaxf(acc[j], 0.0f);   // ReLU
        }
    }
    __syncthreads();

    // ---- GEMM2: M-tile pair (p, p+2) shares the B tile -> fragment add ----
    for (int item = wave; item < 64; item += 8) {
        const int p = item & 1;                     // rows 0..31 selector
        const int n = item >> 1;                    // N tile 0..31
        v8f c0 = {}, c1 = {};
        const float* Brow = w2 + (size_t)(n * 16 + l16) * 32;
        const float* A0 = Hs + (p * 16 + l16) * 32;
        const float* A1 = Hs + ((p + 2) * 16 + l16) * 32;
        for (int k0 = 0; k0 < 32; k0 += 4) {
            v2f a0, a1, b;
            a0.x = A0[k0 + 2 * half + 0];  a0.y = A0[k0 + 2 * half + 1];
            a1.x = A1[k0 + 2 * half + 0];  a1.y = A1[k0 + 2 * half + 1];
            b.x  = Brow[k0 + 2 * half + 0]; b.y = Brow[k0 + 2 * half + 1];
            c0 = __builtin_amdgcn_wmma_f32_16x16x4_f32(false, a0, false, b,
                                                       (short)0, c0, false, false);
            c1 = __builtin_amdgcn_wmma_f32_16x16x4_f32(false, a1, false, b,
                                                       (short)0, c1, false, false);
        }
        for (int j = 0; j < 8; ++j) {
            const float s  = c0[j] + c1[j];         // mlp(avg)+mlp(max)
            const float g  = 1.0f / (1.0f + __expf(-s));
            const int   b_ = p * 16 + j + 8 * half; // batch row 0..31
            const int   c_ = n * 16 + l16;          // channel
            att[b_ * 512 + c_] = g;
        }
    }
}

// ---------------------------------------------------------------------------
// Kernel 3: spatial avg/max over channels of (x * ch_att), ch_att in LDS.
// planes[b][0][hw] = avg, planes[b][1][hw] = max.
// ---------------------------------------------------------------------------
__global__ void __launch_bounds__(256) k_spatial_stats(const float* __restrict__ x,
                                                       const float* __restrict__ att,
                                                       float* __restrict__ planes) {
    __shared__ float a[512];
    const int b = blockIdx.y;
    a[threadIdx.x]       = att[b * 512 + threadIdx.x];
    a[threadIdx.x + 256] = att[b * 512 + threadIdx.x + 256];
    __syncthreads();
    const int hw = blockIdx.x * 256 + threadIdx.x;
    if (hw >= 3136) return;
    const float* xp = x + (size_t)b * 512 * 3136 + hw;
    float s = 0.0f, m = NEG_INF;
    for (int c = 0; c < 512; ++c) {
        const float v = xp[(size_t)c * 3136] * a[c];
        s += v;
        m = fmaxf(m, v);
    }
    planes[b * 2 * 3136 + hw]        = s * (1.0f / 512.0f);
    planes[b * 2 * 3136 + 3136 + hw] = m;
}

// ---------------------------------------------------------------------------
// Kernel 4: 7x7 conv (2->1 ch, pad 3) + sigmoid, one block per batch, in LDS.
// ---------------------------------------------------------------------------
__global__ void __launch_bounds__(256) k_conv_sig(const float* __restrict__ planes,
                                                  const float* __restrict__ wc,
                                                  float* __restrict__ satt) {
    __shared__ float P[2 * 3136];
    __shared__ float W[98];
    const int b = blockIdx.x;
    for (int i = threadIdx.x; i < 2 * 3136; i += 256) P[i] = planes[b * 2 * 3136 + i];
    if (threadIdx.x < 98) W[threadIdx.x] = wc[threadIdx.x];
    __syncthreads();
    for (int p = threadIdx.x; p < 3136; p += 256) {
        const int y  = p / 56;
        const int xq = p - y * 56;
        float acc = 0.0f;
        for (int ic = 0; ic < 2; ++ic) {
            for (int ky = 0; ky < 7; ++ky) {
                const int iy = y + ky - 3;
                if (iy < 0 || iy >= 56) continue;
                for (int kx = 0; kx < 7; ++kx) {
                    const int ix = xq + kx - 3;
                    if (ix < 0 || ix >= 56) continue;
                    acc += P[ic * 3136 + iy * 56 + ix] * W[ic * 49 + ky * 7 + kx];
                }
            }
        }
        satt[b * 3136 + p] = 1.0f / (1.0f + __expf(-acc));
    }
}

// ---------------------------------------------------------------------------
// Kernel 5: fused apply: out = x * ch_att[b,c] * sp_att[b,hw], float4 wide.
// ---------------------------------------------------------------------------
__global__ void __launch_bounds__(256) k_apply(const float* __restrict__ x,
                                               const float* __restrict__ att,
                                               const float* __restrict__ satt,
                                               float* __restrict__ out) {
    const int idx = blockIdx.x * 256 + threadIdx.x;     // float4 index
    const int e   = idx * 4;
    const int row = e / 3136;                           // b*512 + c
    const int r   = e - row * 3136;                     // multiple of 4
    const int b   = row >> 9;
    const float ca = att[row];
    const float4 xv = reinterpret_cast<const float4*>(x)[idx];
    const float4 sv = *reinterpret_cast<const float4*>(satt + b * 3136 + r);
    float4 o;
    o.x = xv.x * ca * sv.x;
    o.y = xv.y * ca * sv.y;
    o.z = xv.z * ca * sv.z;
    o.w = xv.w * ca * sv.w;
    reinterpret_cast<float4*>(out)[idx] = o;
}

// ---------------------------------------------------------------------------
extern "C" void kernel_launch(void* const* d_in, const int* in_sizes, int n_in,
                              void* d_out, int out_size, void* d_ws, size_t ws_size,
                              hipStream_t stream) {
    const float* x  = (const float*)d_in[0];   // [32,512,56,56]
    const float* w1 = (const float*)d_in[1];   // [32,512]
    const float* w2 = (const float*)d_in[2];   // [512,32]
    const float* wc = (const float*)d_in[3];   // [1,2,7,7]
    float* out = (float*)d_out;

    float* ws     = (float*)d_ws;
    float* V      = ws;            // 64*512        = 32768 floats
    float* att    = ws + 32768;    // 32*512        = 16384 floats
    float* planes = ws + 49152;    // 32*2*3136     = 200704 floats
    float* satt   = ws + 249856;   // 32*3136       = 100352 floats

    k_channel_pool<<<16384, 256, 0, stream>>>(x, V);
    k_mlp_att<<<1, 256, 0, stream>>>(V, w1, w2, att);
    dim3 g3(13, 32);
    k_spatial_stats<<<g3, 256, 0, stream>>>(x, att, planes);
    k_conv_sig<<<32, 256, 0, stream>>>(planes, wc, satt);
    k_apply<<<50176, 256, 0, stream>>>(x, att, satt, out);   // 12845056 float4
}